// MotionDetectionLifSNN_31516470018556
// MI455X (gfx1250) — compile-verified
//
#include <hip/hip_runtime.h>

#define POOL 4
#define B_  64
#define T_  100
#define H_  128
#define W_  128
#define F_  1024          // (H/4)*(W/4)
#define NN  200           // hidden neurons
#define BT  (B_ * T_)     // 6400 GEMM rows

typedef __attribute__((ext_vector_type(2))) float v2f;
typedef __attribute__((ext_vector_type(8))) float v8f;

// ---------------------------------------------------------------------------
// Kernel 1: 4x4 average pooling.  Streams 419 MB of x once (the HBM-bound
// part of the whole pipeline), writes xp[6400,1024] to workspace (L2-resident).
// One thread per pooled output; adjacent threads read adjacent float4s ->
// fully coalesced 128B transactions.
// ---------------------------------------------------------------------------
__global__ __launch_bounds__(256) void snn_pool_kernel(
    const float* __restrict__ x, float* __restrict__ xp) {
  const int tid = blockIdx.x * 256 + threadIdx.x;     // 0 .. BT*F_-1 exactly
  const int bt  = tid >> 10;                           // / 1024
  const int f   = tid & 1023;
  const int ph  = f >> 5;                              // pooled row
  const int pw  = f & 31;                              // pooled col
  const float* src = x + ((size_t)bt * H_ + ph * 4) * W_ + pw * 4;
  float s = 0.0f;
#pragma unroll
  for (int r = 0; r < 4; ++r) {
    float4 v = *reinterpret_cast<const float4*>(src + (size_t)r * W_);
    s += (v.x + v.y) + (v.z + v.w);
  }
  xp[tid] = s * (1.0f / 16.0f);
}

// ---------------------------------------------------------------------------
// Kernel 2: g1[6400,200] = xp[6400,1024] * W1^T[1024,200] via fp32 WMMA.
// One wave (32 threads) per 16x16 tile; grid = (400 M-tiles, 13 N-tiles).
// Per ISA layouts for V_WMMA_F32_16X16X4_F32:
//   A (16x4):  lane L<16 -> row M=L holds K={k0,k0+1}; lane L>=16 -> K={k0+2,k0+3}
//   B (4x16):  lane L<16 -> col N=L holds K={k0,k0+1}; lane L>=16 -> K={k0+2,k0+3}
//   C/D:       VGPR r, lane L -> (M = r + 8*(L>>4), N = L&15)
// Columns >= 200 are masked to zero with a value select (EXEC stays all-1s).
// ---------------------------------------------------------------------------
__global__ __launch_bounds__(32) void snn_gemm_kernel(
    const float* __restrict__ xp, const float* __restrict__ W1,
    float* __restrict__ g1) {
  const int m0   = blockIdx.x * 16;
  const int lane = threadIdx.x;
  const int lo   = lane & 15;
  const int hi   = lane >> 4;

  const int  col   = blockIdx.y * 16 + lo;
  const int  ccol  = (col < NN) ? col : (NN - 1);
  const float bmask = (col < NN) ? 1.0f : 0.0f;

  const float* arow = xp + (size_t)(m0 + lo) * F_ + 2 * hi;  // 8B aligned
  const float* brow = W1 + (size_t)ccol * F_ + 2 * hi;       // 8B aligned

  v8f acc = {};
#pragma unroll 8
  for (int k0 = 0; k0 < F_; k0 += 4) {
    v2f a = *reinterpret_cast<const v2f*>(arow + k0);
    v2f b = *reinterpret_cast<const v2f*>(brow + k0);
    b.x *= bmask;
    b.y *= bmask;
    acc = __builtin_amdgcn_wmma_f32_16x16x4_f32(
        /*neg_a=*/false, a, /*neg_b=*/false, b,
        /*c_mod=*/(short)0, acc, /*reuse_a=*/false, /*reuse_b=*/false);
  }

  if (col < NN) {
#pragma unroll
    for (int r = 0; r < 8; ++r) {
      const int row = m0 + r + 8 * hi;
      g1[(size_t)row * NN + col] = acc[r];
    }
  }
}

// ---------------------------------------------------------------------------
// Kernel 3: LIF scan (beta1=0.95, subtract-reset, thr=1) + GEMV with W2 +
// LI integrator (beta2=0.9, thr=1e5 never fires).  One block per batch,
// thread o owns neuron o's membrane in a register; per-step LDS reduction.
// ---------------------------------------------------------------------------
__global__ __launch_bounds__(256) void snn_scan_kernel(
    const float* __restrict__ g1, const float* __restrict__ W2,
    float* __restrict__ y) {
  __shared__ float sdata[256];
  const int b = blockIdx.x;
  const int o = threadIdx.x;
  const float w2  = (o < NN) ? W2[o] : 0.0f;
  float mem  = 0.0f;   // LIF membrane (per neuron)
  float ymem = 0.0f;   // LI membrane (thread 0 only)

  for (int t = 0; t < T_; ++t) {
    const float g = (o < NN) ? g1[(size_t)(b * T_ + t) * NN + o] : 0.0f;
    const float reset = (mem > 1.0f) ? 1.0f : 0.0f;   // uses previous mem
    mem = 0.95f * mem + g - reset;                    // subtract thr=1
    const float z = ((mem - 1.0f) > 0.0f) ? 1.0f : 0.0f;
    sdata[o] = z * w2;                                // w2==0 for o>=200
    __syncthreads();
#pragma unroll
    for (int s = 128; s > 0; s >>= 1) {
      if (o < s) sdata[o] += sdata[o + s];
      __syncthreads();
    }
    if (o == 0) {
      ymem = 0.9f * ymem + sdata[0];                  // LI: thr=1e5 never hit
      y[b * T_ + t] = ymem;
    }
    __syncthreads();                                  // protect sdata[0] read
  }
}

// ---------------------------------------------------------------------------
extern "C" void kernel_launch(void* const* d_in, const int* in_sizes, int n_in,
                              void* d_out, int out_size, void* d_ws, size_t ws_size,
                              hipStream_t stream) {
  const float* x  = (const float*)d_in[0];   // [64,100,128,128]
  const float* W1 = (const float*)d_in[1];   // [200,1024]
  const float* W2 = (const float*)d_in[2];   // [1,200]
  float* y = (float*)d_out;                  // [64,100,1]

  float* xp = (float*)d_ws;                                          // 26.2 MB
  float* g1 = (float*)((char*)d_ws + (size_t)BT * F_ * sizeof(float)); // 5.1 MB

  snn_pool_kernel<<<(BT * F_) / 256, 256, 0, stream>>>(x, xp);
  snn_gemm_kernel<<<dim3(BT / 16, 13), 32, 0, stream>>>(xp, W1, g1);
  snn_scan_kernel<<<B_, 256, 0, stream>>>(g1, W2, y);
}